// PointTransformerFeatureExtractor_29970281791910
// MI455X (gfx1250) — compile-verified
//
#include <hip/hip_runtime.h>
#include <math.h>

typedef __attribute__((ext_vector_type(2))) float v2f;
typedef __attribute__((ext_vector_type(8))) float v8f;

#define B_   8
#define N_   4096
#define KNN_ 16

// ---------------------------------------------------------------------------
// Normalize: center per batch, scale by max radius.  pos[b,n,3]
// ---------------------------------------------------------------------------
__global__ __launch_bounds__(256)
void normalize_kernel(const float* __restrict__ x, float* __restrict__ pos, int N)
{
    __shared__ float r0[256], r1[256], r2[256];
    __shared__ float meanv[3];
    __shared__ float invs;
    int b = blockIdx.x;
    int tid = threadIdx.x;
    const float* xb = x + (size_t)b * 3 * N;
    float s0 = 0.f, s1 = 0.f, s2 = 0.f;
    for (int n = tid; n < N; n += 256) { s0 += xb[n]; s1 += xb[N + n]; s2 += xb[2 * N + n]; }
    r0[tid] = s0; r1[tid] = s1; r2[tid] = s2; __syncthreads();
    for (int s = 128; s > 0; s >>= 1) {
        if (tid < s) { r0[tid] += r0[tid + s]; r1[tid] += r1[tid + s]; r2[tid] += r2[tid + s]; }
        __syncthreads();
    }
    if (tid == 0) { meanv[0] = r0[0] / N; meanv[1] = r1[0] / N; meanv[2] = r2[0] / N; }
    __syncthreads();
    float m0 = meanv[0], m1 = meanv[1], m2 = meanv[2];
    float mx = 0.f;
    for (int n = tid; n < N; n += 256) {
        float dx = xb[n] - m0, dy = xb[N + n] - m1, dz = xb[2 * N + n] - m2;
        mx = fmaxf(mx, dx * dx + dy * dy + dz * dz);
    }
    r0[tid] = mx; __syncthreads();
    for (int s = 128; s > 0; s >>= 1) {
        if (tid < s) r0[tid] = fmaxf(r0[tid], r0[tid + s]);
        __syncthreads();
    }
    if (tid == 0) invs = 1.0f / (sqrtf(r0[0]) + 1e-8f);
    __syncthreads();
    float inv = invs;
    for (int n = tid; n < N; n += 256) {
        float* pp = pos + ((size_t)b * N + n) * 3;
        pp[0] = (xb[n] - m0) * inv;
        pp[1] = (xb[N + n] - m1) * inv;
        pp[2] = (xb[2 * N + n] - m2) * inv;
    }
}

// ---------------------------------------------------------------------------
// KNN: whole batch's point cloud in LDS (48KB << 320KB/WGP); per-thread
// register insertion-sort of top-16 by -dist2, stable ties (matches top_k).
// ---------------------------------------------------------------------------
__global__ __launch_bounds__(256)
void knn_kernel(const float* __restrict__ pos, int* __restrict__ knn, int N)
{
    __shared__ float sp[N_ * 3];
    int b = blockIdx.y;
    int tid = threadIdx.x;
    const float* pb = pos + (size_t)b * N * 3;
    for (int i = tid; i < N * 3; i += 256) sp[i] = pb[i];
    __syncthreads();
    int n = blockIdx.x * 256 + tid;
    float qx = sp[n * 3], qy = sp[n * 3 + 1], qz = sp[n * 3 + 2];
    float best[KNN_]; int bid[KNN_];
#pragma unroll
    for (int i = 0; i < KNN_; ++i) { best[i] = -3.0e38f; bid[i] = 0; }
    for (int m = 0; m < N; ++m) {
        float dx = qx - sp[m * 3], dy = qy - sp[m * 3 + 1], dz = qz - sp[m * 3 + 2];
        float nd = -(dx * dx + dy * dy + dz * dz);
        if (nd > best[KNN_ - 1]) {
            best[KNN_ - 1] = nd; bid[KNN_ - 1] = m;
#pragma unroll
            for (int i = KNN_ - 1; i > 0; --i) {
                if (best[i] > best[i - 1]) {
                    float tf = best[i]; best[i] = best[i - 1]; best[i - 1] = tf;
                    int ti = bid[i]; bid[i] = bid[i - 1]; bid[i - 1] = ti;
                }
            }
        }
    }
    int* out = knn + ((size_t)b * N + n) * KNN_;
#pragma unroll
    for (int i = 0; i < KNN_; ++i) out[i] = bid[i];
}

// ---------------------------------------------------------------------------
// feat0 = pos @ in_w(3x16) + in_b   (tiny K=3: VALU)
// ---------------------------------------------------------------------------
__global__ __launch_bounds__(256)
void input_proj_kernel(const float* __restrict__ pos, const float* __restrict__ w,
                       const float* __restrict__ bias, float* __restrict__ feat, int BN)
{
    int i = blockIdx.x * 256 + threadIdx.x;
    if (i >= BN) return;
    float p0 = pos[i * 3], p1 = pos[i * 3 + 1], p2 = pos[i * 3 + 2];
#pragma unroll
    for (int o = 0; o < 16; ++o)
        feat[(size_t)i * 16 + o] = p0 * w[o] + p1 * w[16 + o] + p2 * w[32 + o] + bias[o];
}

// ---------------------------------------------------------------------------
// WMMA f32 GEMM: C[M,N] = A[M,Kd] @ W[Kd,N] + bias, optional relu.
// NT = N/16 (compile-time).  Each wave owns one 16-row M-tile and ALL NT
// column tiles (NT v8f accumulators stay in registers): the streamed A
// matrix is read exactly once per GEMM (one b64 per k-step, shared by NT
// WMMAs); weights are tiny and cache-resident.  4 waves per block.
// A 16x4 layout: lane&15 = M row, lanes>=16 hold K+2/K+3 (v2f per lane).
// C/D v8f: vgpr j holds M=j (lanes 0-15) / M=j+8 (lanes 16-31), N=lane&15.
// ---------------------------------------------------------------------------
template <int NT>
__global__ __launch_bounds__(128)
void wmma_gemm_nt_kernel(const float* __restrict__ A, const float* __restrict__ W,
                         const float* __restrict__ bias, float* __restrict__ C,
                         int Kd, int relu)
{
    constexpr int N = NT * 16;
    int wave = threadIdx.x >> 5;
    int lane = threadIdx.x & 31;
    int mtile = blockIdx.x * 4 + wave;
    int row = mtile * 16 + (lane & 15);
    int colb = lane & 15;
    int khalf = (lane >> 4) * 2;
    v8f acc[NT] = {};
    for (int k = 0; k < Kd; k += 4) {
        v2f a;
        a.x = A[(size_t)row * Kd + k + khalf];
        a.y = A[(size_t)row * Kd + k + khalf + 1];
#pragma unroll
        for (int nt = 0; nt < NT; ++nt) {
            v2f b;
            b.x = W[(size_t)(k + khalf) * N + nt * 16 + colb];
            b.y = W[(size_t)(k + khalf + 1) * N + nt * 16 + colb];
            acc[nt] = __builtin_amdgcn_wmma_f32_16x16x4_f32(false, a, false, b, (short)0,
                                                            acc[nt], false, false);
        }
    }
    int rbase = mtile * 16 + ((lane >> 4) * 8);
#pragma unroll
    for (int nt = 0; nt < NT; ++nt) {
        int col = nt * 16 + colb;
        float bv = bias[col];
#pragma unroll
        for (int j = 0; j < 8; ++j) {
            float v = acc[nt][j] + bv;
            if (relu) v = fmaxf(v, 0.0f);
            C[(size_t)(rbase + j) * N + col] = v;
        }
    }
}

static void launch_gemm(const float* A, const float* W, const float* bias, float* C,
                        int M, int Nc, int Kd, int relu, hipStream_t stream)
{
    dim3 grid(M / 64);  // 4 waves/block, one 16-row tile per wave
    switch (Nc) {
        case 16:  wmma_gemm_nt_kernel<1><<<grid, 128, 0, stream>>>(A, W, bias, C, Kd, relu); break;
        case 32:  wmma_gemm_nt_kernel<2><<<grid, 128, 0, stream>>>(A, W, bias, C, Kd, relu); break;
        case 64:  wmma_gemm_nt_kernel<4><<<grid, 128, 0, stream>>>(A, W, bias, C, Kd, relu); break;
        default:  wmma_gemm_nt_kernel<8><<<grid, 128, 0, stream>>>(A, W, bias, C, Kd, relu); break;
    }
}

// ---------------------------------------------------------------------------
// h1 = relu(pos_diff @ pm1_w(3xC) + pm1_b), one batch; rows = N*K
// ---------------------------------------------------------------------------
__global__ __launch_bounds__(256)
void pe_mlp1_kernel(const float* __restrict__ pos, const int* __restrict__ knn,
                    const float* __restrict__ w, const float* __restrict__ bias,
                    float* __restrict__ h1, int b, int N, int C)
{
    __shared__ float sw[4 * 128];
    for (int i = threadIdx.x; i < 3 * C; i += 256) sw[i] = w[i];
    for (int i = threadIdx.x; i < C; i += 256) sw[3 * C + i] = bias[i];
    __syncthreads();
    int r = blockIdx.x * 256 + threadIdx.x;
    if (r >= N * KNN_) return;
    int n = r >> 4, k = r & 15;
    int j = knn[((size_t)b * N + n) * KNN_ + k];
    const float* pb = pos + (size_t)b * N * 3;
    float dx = pb[n * 3] - pb[j * 3];
    float dy = pb[n * 3 + 1] - pb[j * 3 + 1];
    float dz = pb[n * 3 + 2] - pb[j * 3 + 2];
    float* hr = h1 + (size_t)r * C;
    for (int c = 0; c < C; ++c)
        hr[c] = fmaxf(dx * sw[c] + dy * sw[C + c] + dz * sw[2 * C + c] + sw[3 * C + c], 0.f);
}

// ---------------------------------------------------------------------------
// h = fc - fn_gathered + pe ; fnpe = fn_gathered + pe  (one batch)
// fnpe written over pe in-place (same-index read-then-write per thread: safe)
// ---------------------------------------------------------------------------
__global__ __launch_bounds__(256)
void combine_kernel(const float* __restrict__ fc_all, const float* __restrict__ fn_all,
                    const float* __restrict__ pe, const int* __restrict__ knn,
                    float* __restrict__ h, float* __restrict__ fnpe,
                    int b, int N, int C)
{
    size_t total = (size_t)N * KNN_ * C;
    size_t t = (size_t)blockIdx.x * 256 + threadIdx.x;
    if (t >= total) return;
    int c = (int)(t % C);
    size_t r = t / C;
    int n = (int)(r >> 4), k = (int)(r & 15);
    int j = knn[((size_t)b * N + n) * KNN_ + k];
    float fn = fn_all[((size_t)b * N + j) * C + c];
    float p = pe[t];
    float fcv = fc_all[((size_t)b * N + n) * C + c];
    fnpe[t] = fn + p;
    h[t] = fcv - fn + p;
}

// ---------------------------------------------------------------------------
// s = a @ s_w + s_b ; softmax over K ; agg = sum_k w*(fn+pe) ; out = fc + agg
// writes next-layer feat and the concat slice.  One thread per point.
// ---------------------------------------------------------------------------
__global__ __launch_bounds__(64)
void attn_out_kernel(const float* __restrict__ a, const float* __restrict__ fnpe,
                     const float* __restrict__ fc_all, const float* __restrict__ sw,
                     const float* __restrict__ sb, float* __restrict__ feat_out,
                     float* __restrict__ cat, int b, int N, int C,
                     int cat_off, int cat_stride)
{
    int n = blockIdx.x * 64 + threadIdx.x;
    if (n >= N) return;
    float s[KNN_];
    float smax = -3.0e38f;
#pragma unroll
    for (int k = 0; k < KNN_; ++k) {
        const float* ar = a + ((size_t)n * KNN_ + k) * C;
        float acc = sb[0];
        for (int c = 0; c < C; ++c) acc += ar[c] * sw[c];
        s[k] = acc;
        smax = fmaxf(smax, acc);
    }
    float denom = 0.f;
#pragma unroll
    for (int k = 0; k < KNN_; ++k) { s[k] = expf(s[k] - smax); denom += s[k]; }
    float inv = 1.0f / denom;
    for (int c = 0; c < C; ++c) {
        float agg = 0.f;
#pragma unroll
        for (int k = 0; k < KNN_; ++k) agg += s[k] * fnpe[((size_t)n * KNN_ + k) * C + c];
        float outv = fc_all[((size_t)b * N + n) * C + c] + agg * inv;
        feat_out[((size_t)b * N + n) * C + c] = outv;
        cat[((size_t)b * N + n) * cat_stride + cat_off + c] = outv;
    }
}

// ---------------------------------------------------------------------------
// Global pooling: g[b, 0:240] = max over n; g[b, 240:480] = mean over n
// ---------------------------------------------------------------------------
__global__ __launch_bounds__(256)
void pool_kernel(const float* __restrict__ cat, float* __restrict__ g, int B, int N, int CC)
{
    int t = blockIdx.x * 256 + threadIdx.x;
    if (t >= B * CC) return;
    int b = t / CC, ch = t % CC;
    const float* p = cat + (size_t)b * N * CC + ch;
    float mx = -3.0e38f, sm = 0.f;
    for (int n = 0; n < N; ++n) {
        float v = p[(size_t)n * CC];
        mx = fmaxf(mx, v);
        sm += v;
    }
    g[b * 2 * CC + ch] = mx;
    g[b * 2 * CC + CC + ch] = sm / N;
}

// ---------------------------------------------------------------------------
// Final FC: out[b,o] = g[b,:] @ fc_w(480x256) + fc_b   (M=8, VALU)
// ---------------------------------------------------------------------------
__global__ __launch_bounds__(256)
void final_fc_kernel(const float* __restrict__ g, const float* __restrict__ w,
                     const float* __restrict__ bias, float* __restrict__ out,
                     int B, int In, int Out)
{
    int t = blockIdx.x * 256 + threadIdx.x;
    if (t >= B * Out) return;
    int b = t / Out, o = t % Out;
    float acc = bias[o];
    for (int i = 0; i < In; ++i) acc += g[b * In + i] * w[(size_t)i * Out + o];
    out[t] = acc;
}

// ---------------------------------------------------------------------------
static inline char* carve(char*& p, size_t bytes)
{
    char* r = p;
    p += (bytes + 255) & ~(size_t)255;
    return r;
}

extern "C" void kernel_launch(void* const* d_in, const int* in_sizes, int n_in,
                              void* d_out, int out_size, void* d_ws, size_t ws_size,
                              hipStream_t stream)
{
    (void)in_sizes; (void)n_in; (void)out_size; (void)ws_size;
    const int B = B_, N = N_;
    const int BN = B * N;
    const int Cin[4]  = {16, 16, 32, 64};
    const int Cout[4] = {16, 32, 64, 128};

    // inputs in setup_inputs() dict (insertion) order, leaves flattened
    const float* x    = (const float*)d_in[0];
    const float* in_w = (const float*)d_in[1];
    const float* in_b = (const float*)d_in[2];
    struct LayerP {
        const float *c_w, *c_b, *n_w, *n_b, *pm1_w, *pm1_b, *pm2_w, *pm2_b,
                    *am1_w, *am1_b, *am2_w, *am2_b, *s_w, *s_b;
    } L[4];
    int p = 3;
    for (int l = 0; l < 4; ++l) {
        L[l].c_w   = (const float*)d_in[p++]; L[l].c_b   = (const float*)d_in[p++];
        L[l].n_w   = (const float*)d_in[p++]; L[l].n_b   = (const float*)d_in[p++];
        L[l].pm1_w = (const float*)d_in[p++]; L[l].pm1_b = (const float*)d_in[p++];
        L[l].pm2_w = (const float*)d_in[p++]; L[l].pm2_b = (const float*)d_in[p++];
        L[l].am1_w = (const float*)d_in[p++]; L[l].am1_b = (const float*)d_in[p++];
        L[l].am2_w = (const float*)d_in[p++]; L[l].am2_b = (const float*)d_in[p++];
        L[l].s_w   = (const float*)d_in[p++]; L[l].s_b   = (const float*)d_in[p++];
    }
    const float* fc_w = (const float*)d_in[p++];
    const float* fc_b = (const float*)d_in[p++];

    // workspace carve (~202 MB)
    char* wsp = (char*)d_ws;
    float* pos    = (float*)carve(wsp, sizeof(float) * BN * 3);
    int*   knn    = (int*)  carve(wsp, sizeof(int) * BN * KNN_);
    float* featA  = (float*)carve(wsp, sizeof(float) * (size_t)BN * 128);
    float* featB  = (float*)carve(wsp, sizeof(float) * (size_t)BN * 128);
    float* fc_all = (float*)carve(wsp, sizeof(float) * (size_t)BN * 128);
    float* fn_all = (float*)carve(wsp, sizeof(float) * (size_t)BN * 128);
    float* buf1   = (float*)carve(wsp, sizeof(float) * (size_t)N * KNN_ * 128);
    float* buf2   = (float*)carve(wsp, sizeof(float) * (size_t)N * KNN_ * 128);
    float* buf3   = (float*)carve(wsp, sizeof(float) * (size_t)N * KNN_ * 128);
    float* cat    = (float*)carve(wsp, sizeof(float) * (size_t)BN * 240);
    float* g      = (float*)carve(wsp, sizeof(float) * B * 480);

    normalize_kernel<<<B, 256, 0, stream>>>(x, pos, N);
    knn_kernel<<<dim3(N / 256, B), 256, 0, stream>>>(pos, knn, N);
    input_proj_kernel<<<(BN + 255) / 256, 256, 0, stream>>>(pos, in_w, in_b, featA, BN);

    float* fin = featA;
    float* fout = featB;
    int cat_off = 0;
    for (int l = 0; l < 4; ++l) {
        int Ci = Cin[l], Co = Cout[l];
        // fc_all = feat @ c_w + c_b ; fn_all = feat @ n_w + n_b   [BN, Co]
        launch_gemm(fin, L[l].c_w, L[l].c_b, fc_all, BN, Co, Ci, 0, stream);
        launch_gemm(fin, L[l].n_w, L[l].n_b, fn_all, BN, Co, Ci, 0, stream);
        const int Mr = N * KNN_;  // 65536 rows per batch
        for (int b = 0; b < B; ++b) {
            pe_mlp1_kernel<<<(Mr + 255) / 256, 256, 0, stream>>>(
                pos, knn, L[l].pm1_w, L[l].pm1_b, buf1, b, N, Co);
            launch_gemm(buf1, L[l].pm2_w, L[l].pm2_b, buf2, Mr, Co, Co, 0, stream);  // pe
            size_t total = (size_t)Mr * Co;
            combine_kernel<<<(unsigned)((total + 255) / 256), 256, 0, stream>>>(
                fc_all, fn_all, buf2, knn, buf1, buf2, b, N, Co);                    // h, fnpe
            launch_gemm(buf1, L[l].am1_w, L[l].am1_b, buf3, Mr, Co, Co, 1, stream);  // relu
            launch_gemm(buf3, L[l].am2_w, L[l].am2_b, buf1, Mr, Co, Co, 0, stream);  // a
            attn_out_kernel<<<(N + 63) / 64, 64, 0, stream>>>(
                buf1, buf2, fc_all, L[l].s_w, L[l].s_b, fout, cat, b, N, Co, cat_off, 240);
        }
        cat_off += Co;
        float* tmp = fin; fin = fout; fout = tmp;
    }

    pool_kernel<<<(B * 240 + 255) / 256, 256, 0, stream>>>(cat, g, B, N, 240);
    final_fc_kernel<<<(B * 256 + 255) / 256, 256, 0, stream>>>(g, fc_w, fc_b,
                                                               (float*)d_out, B, 480, 256);
}